// Ouptut_Analyzer_25778393711223
// MI455X (gfx1250) — compile-verified
//
#include <hip/hip_runtime.h>
#include <hip/hip_bf16.h>

#define EPSF 1e-8f
#define BD 256
#define SD 13
#define VD 18
#define DD 234      // S*V
#define KP 240      // padded K
#define WW 41000    // total words
#define NWORDS 40000
#define PD 42

typedef __attribute__((ext_vector_type(2))) float v2f;
typedef __attribute__((ext_vector_type(8))) float v8f;

// ---- order-preserving float->u32 key (max-reduction friendly) ----
__device__ __forceinline__ unsigned int fkey(float f) {
  unsigned int u = __float_as_uint(f);
  return (u & 0x80000000u) ? ~u : (u | 0x80000000u);
}
// pack key in high 32, (~col) in low 32 so ties resolve to the LOWEST column
__device__ __forceinline__ unsigned long long pkey(float f, int col) {
  return ((unsigned long long)fkey(f) << 32) |
         (unsigned long long)(0xFFFFFFFFu - (unsigned int)col);
}
__device__ __forceinline__ unsigned long long shfl_xor_u64(unsigned long long v, int m) {
  unsigned int lo = __shfl_xor((unsigned int)v, m, 32);
  unsigned int hi = __shfl_xor((unsigned int)(v >> 32), m, 32);
  return ((unsigned long long)hi << 32) | lo;
}
__device__ __forceinline__ unsigned long long maxu64(unsigned long long a, unsigned long long b) {
  return a > b ? a : b;
}

// exact fp32 WMMA: D = A(16x4) * B(4x16) + C(16x16)
__device__ __forceinline__ v8f wmma_k4(v2f a, v2f b, v8f c) {
  return __builtin_amdgcn_wmma_f32_16x16x4_f32(false, a, false, b, (short)0, c, false, false);
}

// ---------------- kernel 0: zero the packed-key arrays ----------------
__global__ void zero_keys_kernel(unsigned long long* __restrict__ keys) {
  int t = blockIdx.x * blockDim.x + threadIdx.x;
  if (t < 3 * BD) keys[t] = 0ull;
}

// ---------------- kernel 1: per-row precompute ----------------
// a1mat[b][d] = x (K-padded) ; a2mat[b][d] = l1-l2 (K-padded)
// xsq[b] = sum x^2 ; l2sum[b] = sum log(1-x+eps)
__global__ __launch_bounds__(32) void pre_kernel(const float* __restrict__ inputs,
                                                 float* __restrict__ a1mat,
                                                 float* __restrict__ a2mat,
                                                 float* __restrict__ xsqA,
                                                 float* __restrict__ l2sA) {
  const int b = blockIdx.x, lane = threadIdx.x;
  float xs = 0.f, l2s = 0.f;
  for (int d = lane; d < KP; d += 32) {
    float a1 = 0.f, a2 = 0.f;
    if (d < DD) {
      float x = inputs[(size_t)b * DD + d];
      a1 = x;
      xs += x * x;
      float l1 = logf(x + EPSF);
      float l2 = logf(1.f - x + EPSF);
      l2s += l2;
      a2 = l1 - l2;
    }
    a1mat[(size_t)b * KP + d] = a1;
    a2mat[(size_t)b * KP + d] = a2;
  }
  for (int m = 16; m >= 1; m >>= 1) {
    xs += __shfl_xor(xs, m, 32);
    l2s += __shfl_xor(l2s, m, 32);
  }
  if (lane == 0) { xsqA[b] = xs; l2sA[b] = l2s; }
}

// ---------------- kernel 2: phoneme argmax per (b,s) ----------------
__global__ void phoneme_kernel(const float* __restrict__ inputs,
                               const float* __restrict__ ph,
                               float* __restrict__ out) {
  int t = blockIdx.x * blockDim.x + threadIdx.x;
  if (t >= BD * SD) return;
  const float* xv = inputs + (size_t)t * VD;
  int best = 0;
  float bestv = -3.0e38f;
  for (int p = 0; p < PD; ++p) {
    const float* pv = ph + p * VD;
    float dot = 0.f, ps = 0.f;
#pragma unroll
    for (int v = 0; v < VD; ++v) { dot += xv[v] * pv[v]; ps += pv[v] * pv[v]; }
    float sc = dot * rsqrtf(ps);  // step-norm is a positive per-thread constant: argmax unchanged
    if (sc > bestv) { bestv = sc; best = p; }
  }
  out[2304 + t] = (float)best;
}

// ---------------- kernel 3: fused dual-GEMM + argmax/argmin reduction ----------------
// grid.x tiles W in strips of 128 (8 waves x 16 cols); grid.y tiles B in strips of 32
__global__ __launch_bounds__(256) void gemm_kernel(const float* __restrict__ a1mat,
                                                   const float* __restrict__ allw,
                                                   const float* __restrict__ added,
                                                   const float* __restrict__ a2mat,
                                                   const float* __restrict__ xsqA,
                                                   unsigned long long* __restrict__ keys) {
  __shared__ float lA1[32 * KP];                 // x rows (padded)
  __shared__ float lA2[32 * KP];                 // (l1-l2) rows (padded)
  __shared__ unsigned long long redK[96];        // 3 metrics x 32 local rows
  __shared__ float lXsq[32];

  const int tid = threadIdx.x;
  const int yBase = blockIdx.y * 32;
  if (tid < 96) redK[tid] = 0ull;
  if (tid < 32) lXsq[tid] = xsqA[yBase + tid];

  // ---- stage both A tiles with async global->LDS DMA (ASYNCcnt-tracked) ----
  {
    const unsigned ldsA1 = (unsigned)(size_t)(void*)lA1;  // low 32 bits of generic addr = LDS offset
    const unsigned ldsA2 = (unsigned)(size_t)(void*)lA2;
    // 2 matrices x 32 rows x 60 b128-chunks = 3840 chunks; 15 async instrs per wave
    for (int i = tid; i < 2 * 32 * 60; i += 256) {
      const int mat = i / (32 * 60);
      const int rem = i - mat * (32 * 60);
      const int row = rem / 60;
      const int c4 = (rem - row * 60) * 4;
      const float* gsrc = (mat ? a2mat : a1mat) + (size_t)(yBase + row) * KP + c4;
      const unsigned loff = (mat ? ldsA2 : ldsA1) + (unsigned)((row * KP + c4) * 4);
      const unsigned long long ga = (unsigned long long)(size_t)gsrc;
      asm volatile("global_load_async_to_lds_b128 %0, %1, off"
                   :: "v"(loff), "v"(ga) : "memory");
    }
    asm volatile("s_wait_asynccnt 0x0" ::: "memory");
  }
  __syncthreads();

  const int lane = tid & 31;
  const int wave = tid >> 5;
  const int h = lane >> 4;       // which half-wave (selects K pair)
  const int c16 = lane & 15;     // column within 16-wide strip
  const int col = blockIdx.x * 128 + wave * 16 + c16;
  const bool colOK = (col < WW);
  const int colC = colOK ? col : (WW - 1);       // clamp: OOB lanes read a valid row,
                                                 // their results are discarded via keys
  const float* wrow = (colC < NWORDS) ? (allw + (size_t)colC * DD)
                                      : (added + (size_t)(colC - NWORDS) * DD);
  __builtin_prefetch(wrow, 0, 3);
  __builtin_prefetch(wrow + 128, 0, 3);

  v8f d0 = {}, d1 = {}, c0 = {}, c1 = {};
  float wsqp = 0.f;
  const float* pa1 = lA1 + c16 * KP;
  const float* pa2 = lA2 + c16 * KP;

#pragma unroll 4
  for (int k0 = 0; k0 < 232; k0 += 4) {
    const int kk = k0 + 2 * h;
    const v2f bv = *(const v2f*)(wrow + kk);     // unconditional: no exec-mask churn
    wsqp += bv.x * bv.x + bv.y * bv.y;
    v2f a10 = *(const v2f*)(pa1 + kk);
    v2f a11 = *(const v2f*)(pa1 + 16 * KP + kk);
    v2f a20 = *(const v2f*)(pa2 + kk);
    v2f a21 = *(const v2f*)(pa2 + 16 * KP + kk);
    d0 = wmma_k4(a10, bv, d0);
    d1 = wmma_k4(a11, bv, d1);
    c0 = wmma_k4(a20, bv, c0);
    c1 = wmma_k4(a21, bv, c1);
  }
  {  // K tail: k = 232..233 valid, 234..235 must be zero (LDS side is zero-padded)
    const int kk = 232 + 2 * h;
    v2f bv; bv.x = 0.f; bv.y = 0.f;
    if (h == 0) bv = *(const v2f*)(wrow + kk);
    wsqp += bv.x * bv.x + bv.y * bv.y;
    v2f a10 = *(const v2f*)(pa1 + kk);
    v2f a11 = *(const v2f*)(pa1 + 16 * KP + kk);
    v2f a20 = *(const v2f*)(pa2 + kk);
    v2f a21 = *(const v2f*)(pa2 + 16 * KP + kk);
    d0 = wmma_k4(a10, bv, d0);
    d1 = wmma_k4(a11, bv, d1);
    c0 = wmma_k4(a20, bv, c0);
    c1 = wmma_k4(a21, bv, c1);
  }

  // full ||w||^2 for this column: combine the two half-wave partials
  const float wsqf = wsqp + __shfl_xor(wsqp, 16, 32);

#pragma unroll
  for (int t = 0; t < 2; ++t) {
    const v8f dacc = t ? d1 : d0;
    const v8f cacc = t ? c1 : c0;
#pragma unroll
    for (int r = 0; r < 8; ++r) {
      const int brow = t * 16 + r + 8 * h;  // local B row this lane's value belongs to
      const float xs = lXsq[brow];
      const float dotv = dacc[r];
      const float cedv = cacc[r];
      unsigned long long kcs = 0ull, ksq = 0ull, kce = 0ull;
      if (colOK) {
        const float cs = dotv / (sqrtf(wsqf) * sqrtf(xs));
        const float sq = fmaxf(xs + wsqf - 2.f * dotv, 0.f);
        kcs = pkey(cs, col);     // argmax cosine
        ksq = pkey(-sq, col);    // argmin sq-dist (== argmin mse == argmin euclid)
        kce = pkey(cedv, col);   // argmax ce-dot  (== argmin ce)
      }
#pragma unroll
      for (int m = 8; m >= 1; m >>= 1) {
        kcs = maxu64(kcs, shfl_xor_u64(kcs, m));
        ksq = maxu64(ksq, shfl_xor_u64(ksq, m));
        kce = maxu64(kce, shfl_xor_u64(kce, m));
      }
      if (c16 == 0) {
        atomicMax(&redK[brow], kcs);
        atomicMax(&redK[32 + brow], ksq);
        atomicMax(&redK[64 + brow], kce);
      }
    }
  }
  __syncthreads();
  if (tid < 32) {
    const int g = yBase + tid;
    atomicMax(&keys[g], redK[tid]);
    atomicMax(&keys[256 + g], redK[32 + tid]);
    atomicMax(&keys[512 + g], redK[64 + tid]);
  }
}

// ---------------- kernel 4: per-row finalize (values at idx + all accuracy flags) ----------------
__global__ __launch_bounds__(32) void row_kernel(const float* __restrict__ inputs,
                                                 const int* __restrict__ widx,
                                                 const int* __restrict__ phidx,
                                                 const float* __restrict__ added,
                                                 const float* __restrict__ allw,
                                                 const float* __restrict__ xsqA,
                                                 const float* __restrict__ l2sA,
                                                 const unsigned long long* __restrict__ keys,
                                                 float* __restrict__ out) {
  const int b = blockIdx.x, lane = threadIdx.x;
  const int idx = widx[b];
  const float* wrow = (idx < NWORDS) ? (allw + (size_t)idx * DD)
                                     : (added + (size_t)(idx - NWORDS) * DD);
  const float* xrow = inputs + (size_t)b * DD;
  float dot = 0.f, wsq = 0.f, ced = 0.f;
  for (int d = lane; d < DD; d += 32) {
    float x = xrow[d], w = wrow[d];
    dot += x * w;
    wsq += w * w;
    float a2 = logf(x + EPSF) - logf(1.f - x + EPSF);
    ced += a2 * w;
  }
  for (int m = 16; m >= 1; m >>= 1) {
    dot += __shfl_xor(dot, m, 32);
    wsq += __shfl_xor(wsq, m, 32);
    ced += __shfl_xor(ced, m, 32);
  }
  if (lane == 0) {
    const float xs = xsqA[b], l2s = l2sA[b];
    const float sq = fmaxf(xs + wsq - 2.f * dot, 0.f);
    out[b]       = dot / (sqrtf(wsq) * sqrtf(xs));
    out[256 + b] = sq / (float)DD;
    out[512 + b] = sqrtf(sq);
    out[768 + b] = -(ced + l2s) / (float)DD;
    const int acs = (int)(0xFFFFFFFFu - (unsigned int)(keys[b]       & 0xFFFFFFFFull));
    const int asq = (int)(0xFFFFFFFFu - (unsigned int)(keys[256 + b] & 0xFFFFFFFFull));
    const int ace = (int)(0xFFFFFFFFu - (unsigned int)(keys[512 + b] & 0xFFFFFFFFull));
    out[1024 + b] = (acs == idx) ? 1.f : 0.f;
    out[1280 + b] = (asq == idx) ? 1.f : 0.f;
    out[1536 + b] = (asq == idx) ? 1.f : 0.f;
    out[1792 + b] = (ace == idx) ? 1.f : 0.f;
    float allm = 1.f;
    for (int s = 0; s < SD; ++s)
      if (out[2304 + b * SD + s] != (float)phidx[b * SD + s]) allm = 0.f;
    out[2048 + b] = allm;
  }
}

extern "C" void kernel_launch(void* const* d_in, const int* in_sizes, int n_in,
                              void* d_out, int out_size, void* d_ws, size_t ws_size,
                              hipStream_t stream) {
  const float* inputs = (const float*)d_in[0];
  const int*   widx   = (const int*)d_in[1];
  const int*   phidx  = (const int*)d_in[2];
  const float* added  = (const float*)d_in[3];
  const float* allw   = (const float*)d_in[4];
  const float* ph     = (const float*)d_in[5];
  float* out = (float*)d_out;

  char* ws = (char*)d_ws;
  float* a1mat = (float*)ws;                                // 256*240 floats = 245760 B
  float* a2mat = (float*)(ws + 245760);                     // 256*240 floats = 245760 B
  float* xsqA  = (float*)(ws + 491520);                     // 256 floats
  float* l2sA  = (float*)(ws + 492544);                     // 256 floats
  unsigned long long* keys = (unsigned long long*)(ws + 493568);  // 3*256 u64

  zero_keys_kernel<<<3, 256, 0, stream>>>(keys);
  pre_kernel<<<BD, 32, 0, stream>>>(inputs, a1mat, a2mat, xsqA, l2sA);
  phoneme_kernel<<<(BD * SD + 255) / 256, 256, 0, stream>>>(inputs, ph, out);
  dim3 g((WW + 127) / 128, 8);
  gemm_kernel<<<g, 256, 0, stream>>>(a1mat, allw, added, a2mat, xsqA, keys);
  row_kernel<<<BD, 32, 0, stream>>>(inputs, widx, phidx, added, allw, xsqA, l2sA, keys, out);
}